// GNNRecommender_532575945305
// MI455X (gfx1250) — compile-verified
//
#include <hip/hip_runtime.h>
#include <hip/hip_bf16.h>

typedef __attribute__((ext_vector_type(2))) float v2f;
typedef __attribute__((ext_vector_type(8))) float v8f;

#define FEAT 64

// Native no-return f32 atomic add at device scope (L2 atomic units).
__device__ __forceinline__ void atomic_add_f32_dev(float* p, float v) {
    asm volatile("global_atomic_add_f32 %0, %1, off scope:SCOPE_DEV"
                 :
                 : "v"(p), "v"(v)
                 : "memory");
}

// ---------------- utility kernels ----------------
__global__ void zero_i32_kernel(int* __restrict__ p, int n) {
    int i = blockIdx.x * blockDim.x + threadIdx.x;
    if (i < n) p[i] = 0;
}

// deg[v] += 1 for every incoming edge (dst side). Self-loop added in dinv.
__global__ void deg_count_kernel(const int* __restrict__ dst, int E,
                                 int* __restrict__ deg) {
    int i = blockIdx.x * blockDim.x + threadIdx.x;
    int stride = gridDim.x * blockDim.x;
    for (; i < E; i += stride) atomicAdd(&deg[dst[i]], 1);
}

__global__ void dinv_kernel(const int* __restrict__ deg,
                            float* __restrict__ dinv, int n) {
    int i = blockIdx.x * blockDim.x + threadIdx.x;
    if (i < n) dinv[i] = rsqrtf((float)(deg[i] + 1));  // +1 self-loop, deg>=1
}

// ---------------- GEMM: H = act(A)(nrows x 64) * W(64 x 64), fp32 WMMA ------
// One wave computes one 16x16 output tile (K=64 in 16 steps of 16x16x4).
// Block = 128 threads = 4 waves => one 16-row x 64-col block per workgroup.
// RELU_A fuses the previous layer's ReLU into the A-operand load.
template <int RELU_A>
__global__ __launch_bounds__(128) void gemm_wmma_kernel(
    const float* __restrict__ A, const float* __restrict__ W,
    float* __restrict__ H, int n_rows) {
    const int lane = threadIdx.x & 31;
    const int warp = threadIdx.x >> 5;      // 0..3 -> which 16-col tile
    const int m0   = blockIdx.x * 16;
    const int half = lane >> 4;             // 0: lanes 0-15, 1: lanes 16-31
    const int l15  = lane & 15;
    const int col  = (warp << 4) + l15;     // output column (0..63)

    int row = m0 + l15;
    if (row >= n_rows) row = n_rows - 1;    // clamp (select, keeps EXEC full)
    const float* __restrict__ arow = A + (long long)row * FEAT;

    v8f acc = {0.f, 0.f, 0.f, 0.f, 0.f, 0.f, 0.f, 0.f};

#pragma unroll
    for (int k = 0; k < FEAT; k += 4) {
        const int kk = k + (half << 1);     // this lane-half's K pair
        // A tile 16x4: VGPR0 = K{0|2}, VGPR1 = K{1|3} (contiguous b64 load)
        v2f a;
        a.x = arow[kk];
        a.y = arow[kk + 1];
        if (RELU_A) {
            a.x = fmaxf(a.x, 0.0f);
            a.y = fmaxf(a.y, 0.0f);
        }
        // B tile 4x16: VGPR0 = row K{0|2}, VGPR1 = row K{1|3}, striped over N
        v2f b;
        b.x = W[(long long)kk * FEAT + col];
        b.y = W[(long long)(kk + 1) * FEAT + col];
        acc = __builtin_amdgcn_wmma_f32_16x16x4_f32(
            /*neg_a=*/false, a, /*neg_b=*/false, b,
            /*c_mod=*/(short)0, acc, /*reuse_a=*/false, /*reuse_b=*/false);
    }

    // C/D layout: VGPR i -> row m0+i (lanes 0-15) / m0+8+i (lanes 16-31)
#pragma unroll
    for (int i = 0; i < 8; ++i) {
        int orow = m0 + (half << 3) + i;
        if (orow < n_rows) H[(long long)orow * FEAT + col] = acc[i];
    }
}

// ---- init accumulator with self-loop + bias: acc[i] = dinv[v]^2*h[i]+b[c] --
__global__ void init_acc_kernel(const float* __restrict__ h,
                                const float* __restrict__ dinv,
                                const float* __restrict__ bias,
                                float* __restrict__ acc, int n_rows) {
    long long i = (long long)blockIdx.x * blockDim.x + threadIdx.x;
    const long long total = (long long)n_rows * FEAT;
    const long long stride = (long long)gridDim.x * blockDim.x;
    for (; i < total; i += stride) {
        const int v = (int)(i >> 6);
        const int c = (int)(i & 63);
        const float dv = dinv[v];
        acc[i] = dv * dv * h[i] + bias[c];
    }
}

// ---------------- edge scatter: acc[dst] += dinv[src]*dinv[dst]*h[src] ------
// One wave per edge; each lane handles 2 contiguous columns (coalesced row).
__global__ __launch_bounds__(256) void scatter_kernel(
    const int* __restrict__ src, const int* __restrict__ dst,
    const float* __restrict__ h, const float* __restrict__ dinv,
    float* __restrict__ acc, int E) {
    const int lane   = threadIdx.x & 31;
    int wave         = (blockIdx.x * blockDim.x + threadIdx.x) >> 5;
    const int nwaves = (gridDim.x * blockDim.x) >> 5;

    for (int e = wave; e < E; e += nwaves) {
        const int s = src[e];
        const int d = dst[e];
        const float norm = dinv[s] * dinv[d];
        const float2 hv =
            *(const float2*)(h + (long long)s * FEAT + (lane << 1));
        float* p = acc + (long long)d * FEAT + (lane << 1);
        atomic_add_f32_dev(p, norm * hv.x);
        atomic_add_f32_dev(p + 1, norm * hv.y);
    }
}

// ---------------- launch ----------------
extern "C" void kernel_launch(void* const* d_in, const int* in_sizes, int n_in,
                              void* d_out, int out_size, void* d_ws,
                              size_t ws_size, hipStream_t stream) {
    const float* x  = (const float*)d_in[0];
    const int*   ei = (const int*)d_in[1];   // int32 (JAX canonicalizes int64)
    const float* W1 = (const float*)d_in[2];
    const float* b1 = (const float*)d_in[3];
    const float* W2 = (const float*)d_in[4];
    const float* b2 = (const float*)d_in[5];

    const int N = in_sizes[0] / FEAT;        // 100000 nodes
    const int E = in_sizes[1] / 2;           // 1200000 edges
    const int* src = ei;
    const int* dst = ei + E;

    // workspace carve-up (floats), 1KB-aligned chunks
    const long long NC = (long long)N * FEAT;
    float* ws   = (float*)d_ws;
    float* dinv = ws;                                       // N floats
    float* h    = ws + (((long long)N + 255) / 256) * 256;  // N*64
    float* acc  = h + NC;                                   // N*64
    int*   deg  = (int*)(acc + NC);                         // N ints

    const int mtiles = (N + 15) / 16;        // 6250 (exact)
    const int tb256N = (N + 255) / 256;
    const int tb256E = (E + 255) / 256;

    // degree / normalization (edges only; shared by both layers)
    zero_i32_kernel<<<tb256N, 256, 0, stream>>>(deg, N);
    deg_count_kernel<<<tb256E, 256, 0, stream>>>(dst, E, deg);
    dinv_kernel<<<tb256N, 256, 0, stream>>>(deg, dinv, N);

    // ---- layer 1: acc = GCNConv(x, W1, b1) (pre-ReLU; ReLU fused into L2) --
    gemm_wmma_kernel<0><<<mtiles, 128, 0, stream>>>(x, W1, h, N);
    init_acc_kernel<<<4096, 256, 0, stream>>>(h, dinv, b1, acc, N);
    scatter_kernel<<<2048, 256, 0, stream>>>(src, dst, h, dinv, acc, E);

    // ---- layer 2: d_out = GCNConv(relu(acc), W2, b2) -----------------------
    gemm_wmma_kernel<1><<<mtiles, 128, 0, stream>>>(acc, W2, h, N);
    init_acc_kernel<<<4096, 256, 0, stream>>>(h, dinv, b2, (float*)d_out, N);
    scatter_kernel<<<2048, 256, 0, stream>>>(src, dst, h, dinv, (float*)d_out,
                                             E);
}